// Qwen3MoeSparseMoeBlock_32933809226594
// MI455X (gfx1250) — compile-verified
//
#include <hip/hip_runtime.h>
#include <math.h>

// ---------------- problem constants ----------------
#define T_TOK 1024   // B*S tokens
#define H_DIM 1024   // hidden
#define E_EXP 64     // experts
#define I_DIM 768    // intermediate
#define TOPK 8
#define M_TILE 32    // tokens per workgroup tile (2 WMMA m-tiles)

typedef float v2f __attribute__((ext_vector_type(2)));
typedef float v8f __attribute__((ext_vector_type(8)));

// ---------------- kernel 1: zero output + expert counters ----------------
__global__ __launch_bounds__(256) void moe_init_kernel(float* __restrict__ out,
                                                       int* __restrict__ counts) {
  int i = blockIdx.x * 256 + threadIdx.x;
  if (i < T_TOK * H_DIM) out[i] = 0.0f;
  if (i < E_EXP) counts[i] = 0;
}

// ---------------- kernel 2: router (softmax -> top8 -> renorm -> scatter) ----
__global__ __launch_bounds__(64) void moe_router_kernel(
    const float* __restrict__ x,       // [T,H]
    const float* __restrict__ gate_w,  // [E,H]
    int* __restrict__ counts,          // [E]
    int* __restrict__ tok_list,        // [E,T]
    float* __restrict__ wt_list) {     // [E,T]
  __shared__ float xs[H_DIM];
  __shared__ float lg[E_EXP];
  const int t = blockIdx.x;

  for (int c = threadIdx.x; c < H_DIM; c += 64) xs[c] = x[t * H_DIM + c];
  __syncthreads();

  {
    const int e = threadIdx.x;
    const float* g = gate_w + e * H_DIM;
    float acc = 0.0f;
    for (int h = 0; h < H_DIM; ++h) acc = fmaf(xs[h], g[h], acc);
    lg[e] = acc;
  }
  __syncthreads();

  if (threadIdx.x == 0) {
    float mx = lg[0];
    for (int i = 1; i < E_EXP; ++i) mx = fmaxf(mx, lg[i]);
    float s = 0.0f;
    for (int i = 0; i < E_EXP; ++i) {
      float p = __expf(lg[i] - mx);
      lg[i] = p;
      s += p;
    }
    const float inv = 1.0f / s;
    int idx[TOPK];
    float pw[TOPK];
    float tot = 0.0f;
    for (int k = 0; k < TOPK; ++k) {
      int bi = 0;
      float bv = -1.0f;
      for (int i = 0; i < E_EXP; ++i) {
        float p = lg[i];
        if (p > bv) { bv = p; bi = i; }
      }
      lg[bi] = -1.0f;
      idx[k] = bi;
      pw[k] = bv * inv;
      tot += bv * inv;
    }
    const float rinv = 1.0f / tot;
    for (int k = 0; k < TOPK; ++k) {
      const int e = idx[k];
      const int pos = atomicAdd(&counts[e], 1);
      tok_list[e * T_TOK + pos] = t;
      wt_list[e * T_TOK + pos] = pw[k] * rinv;
    }
  }
}

// ---------------- kernel 3: expert SwiGLU MLP with fp32 WMMA ----------------
// grid = (T_TOK/32 tiles, E_EXP experts), block = 512 (16 waves)
// M-tile = 32 tokens (2 WMMA m-frags share every B fragment: 1 load : 1 wmma)
// phase 1: 48 I-tiles -> 3 per wave x {gate,up} x 2 m  (12 accumulators)
// phase 2: 64 H-tiles -> 4 per wave x 2 m              (8 accumulators)
__global__ __launch_bounds__(512) void moe_expert_kernel(
    const float* __restrict__ x,       // [T,H]
    const float* __restrict__ w_gate,  // [E,H,I]
    const float* __restrict__ w_up,    // [E,H,I]
    const float* __restrict__ w_down,  // [E,I,H]
    const int* __restrict__ counts,
    const int* __restrict__ tok_list,
    const float* __restrict__ wt_list,
    float* __restrict__ out) {         // [T,H] (atomic accumulate)
  __shared__ float smem[M_TILE * H_DIM];  // 128KB: X tile, later act tile (96KB)

  const int e = blockIdx.y;
  const int mt = blockIdx.x;
  const int cnt = counts[e];
  if (mt * M_TILE >= cnt) return;  // block-uniform exit

  const int tid = threadIdx.x;
  const int wv = tid >> 5;   // wave id 0..15
  const int lane = tid & 31;
  const int row = lane & 15; // WMMA M row / N col index
  const int kh = lane >> 4;  // K-pair half held by this lane

  // ---- stage 32 gathered token rows of X into LDS ----
  for (int r = 0; r < M_TILE; ++r) {
    const int li = mt * M_TILE + r;
    const int t = (li < cnt) ? tok_list[e * T_TOK + li] : 0;
    const float* src = x + (size_t)t * H_DIM;
    for (int c = tid; c < H_DIM; c += 512) smem[r * H_DIM + c] = src[c];
  }
  __syncthreads();

  // ---- phase 1: G = X*Wg, U = X*Wu  (K = H_DIM) ----
  const float* wg = w_gate + (size_t)e * H_DIM * I_DIM;
  const float* wu = w_up + (size_t)e * H_DIM * I_DIM;
  const int i0base = wv * 3 * 16;  // 48 cols of I per wave

  v8f aG0[3], aG1[3], aU0[3], aU1[3];
#pragma unroll
  for (int j = 0; j < 3; ++j) {
    aG0[j] = (v8f)0.0f; aG1[j] = (v8f)0.0f;
    aU0[j] = (v8f)0.0f; aU1[j] = (v8f)0.0f;
  }

  for (int k0 = 0; k0 < H_DIM; k0 += 4) {
    const float* ap0 = &smem[row * H_DIM + k0 + 2 * kh];
    const float* ap1 = &smem[(16 + row) * H_DIM + k0 + 2 * kh];
    v2f a0, a1;
    a0.x = ap0[0]; a0.y = ap0[1];
    a1.x = ap1[0]; a1.y = ap1[1];
    const int roff = (k0 + 2 * kh) * I_DIM;
#pragma unroll
    for (int j = 0; j < 3; ++j) {
      const int col = i0base + j * 16 + row;
      v2f bg, bu;
      bg.x = wg[roff + col];
      bg.y = wg[roff + I_DIM + col];
      bu.x = wu[roff + col];
      bu.y = wu[roff + I_DIM + col];
      aG0[j] = __builtin_amdgcn_wmma_f32_16x16x4_f32(
          false, a0, false, bg, (short)0, aG0[j], false, false);
      aG1[j] = __builtin_amdgcn_wmma_f32_16x16x4_f32(
          false, a1, false, bg, (short)0, aG1[j], false, false);
      aU0[j] = __builtin_amdgcn_wmma_f32_16x16x4_f32(
          false, a0, false, bu, (short)0, aU0[j], false, false);
      aU1[j] = __builtin_amdgcn_wmma_f32_16x16x4_f32(
          false, a1, false, bu, (short)0, aU1[j], false, false);
    }
  }

  // ---- epilogue: act = silu(g) * u * routing_weight(row m) ----
  // C layout: (m,n): m = v + 8*kh (+16 for second m-frag), n = row
  float wr0[8], wr1[8];
#pragma unroll
  for (int v = 0; v < 8; ++v) {
    const int l0 = mt * M_TILE + 8 * kh + v;
    const int l1 = l0 + 16;
    wr0[v] = (l0 < cnt) ? wt_list[e * T_TOK + l0] : 0.0f;
    wr1[v] = (l1 < cnt) ? wt_list[e * T_TOK + l1] : 0.0f;
  }
#pragma unroll
  for (int j = 0; j < 3; ++j) {
#pragma unroll
    for (int v = 0; v < 8; ++v) {
      float g0 = aG0[j][v], u0 = aU0[j][v];
      float g1 = aG1[j][v], u1 = aU1[j][v];
      aG0[j][v] = (g0 / (1.0f + __expf(-g0))) * u0 * wr0[v];
      aG1[j][v] = (g1 / (1.0f + __expf(-g1))) * u1 * wr1[v];
    }
  }

  __syncthreads();  // everyone done reading X tile
  // write act tile [32, I_DIM] into same LDS (first 96KB)
#pragma unroll
  for (int j = 0; j < 3; ++j) {
    const int col = i0base + j * 16 + row;
#pragma unroll
    for (int v = 0; v < 8; ++v) {
      const int m = v + 8 * kh;
      smem[m * I_DIM + col] = aG0[j][v];
      smem[(16 + m) * I_DIM + col] = aG1[j][v];
    }
  }
  __syncthreads();

  // ---- phase 2: Out = act * Wd  (K = I_DIM) ----
  const float* wd = w_down + (size_t)e * I_DIM * H_DIM;
  const int h0base = wv * 4 * 16;  // 64 cols of H per wave

  v8f aD0[4], aD1[4];
#pragma unroll
  for (int j = 0; j < 4; ++j) { aD0[j] = (v8f)0.0f; aD1[j] = (v8f)0.0f; }

  for (int k0 = 0; k0 < I_DIM; k0 += 4) {
    const float* ap0 = &smem[row * I_DIM + k0 + 2 * kh];
    const float* ap1 = &smem[(16 + row) * I_DIM + k0 + 2 * kh];
    v2f a0, a1;
    a0.x = ap0[0]; a0.y = ap0[1];
    a1.x = ap1[0]; a1.y = ap1[1];
    const int roff = (k0 + 2 * kh) * H_DIM;
#pragma unroll
    for (int j = 0; j < 4; ++j) {
      const int col = h0base + j * 16 + row;
      v2f b;
      b.x = wd[roff + col];
      b.y = wd[roff + H_DIM + col];
      aD0[j] = __builtin_amdgcn_wmma_f32_16x16x4_f32(
          false, a0, false, b, (short)0, aD0[j], false, false);
      aD1[j] = __builtin_amdgcn_wmma_f32_16x16x4_f32(
          false, a1, false, b, (short)0, aD1[j], false, false);
    }
  }

  // ---- atomic scatter-add into out[token, h] ----
  int tr0[8], tr1[8];
#pragma unroll
  for (int v = 0; v < 8; ++v) {
    const int l0 = mt * M_TILE + 8 * kh + v;
    const int l1 = l0 + 16;
    tr0[v] = (l0 < cnt) ? tok_list[e * T_TOK + l0] : -1;
    tr1[v] = (l1 < cnt) ? tok_list[e * T_TOK + l1] : -1;
  }
#pragma unroll
  for (int j = 0; j < 4; ++j) {
    const int col = h0base + j * 16 + row;
#pragma unroll
    for (int v = 0; v < 8; ++v) {
      if (tr0[v] >= 0) atomicAdd(&out[(size_t)tr0[v] * H_DIM + col], aD0[j][v]);
      if (tr1[v] >= 0) atomicAdd(&out[(size_t)tr1[v] * H_DIM + col], aD1[j][v]);
    }
  }
}

// ---------------- host launch ----------------
extern "C" void kernel_launch(void* const* d_in, const int* in_sizes, int n_in,
                              void* d_out, int out_size, void* d_ws, size_t ws_size,
                              hipStream_t stream) {
  const float* x = (const float*)d_in[0];       // [1,1024,1024]
  const float* gate_w = (const float*)d_in[1];  // [64,1024]
  const float* w_gate = (const float*)d_in[2];  // [64,1024,768]
  const float* w_up = (const float*)d_in[3];    // [64,1024,768]
  const float* w_down = (const float*)d_in[4];  // [64,768,1024]
  float* out = (float*)d_out;

  // workspace: counts[E] | tok_list[E*T] | wt_list[E*T]
  char* ws = (char*)d_ws;
  int* counts = (int*)ws;
  int* tok_list = (int*)(ws + 256);
  float* wt_list = (float*)(ws + 256 + (size_t)E_EXP * T_TOK * sizeof(int));

  moe_init_kernel<<<(T_TOK * H_DIM + 255) / 256, 256, 0, stream>>>(out, counts);
  moe_router_kernel<<<T_TOK, 64, 0, stream>>>(x, gate_w, counts, tok_list, wt_list);
  moe_expert_kernel<<<dim3(T_TOK / M_TILE, E_EXP), 512, 0, stream>>>(
      x, w_gate, w_up, w_down, counts, tok_list, wt_list, out);
}